// SkipGramModel_15032385536593
// MI455X (gfx1250) — compile-verified
//
#include <hip/hip_runtime.h>
#include <hip/hip_bf16.h>

typedef __attribute__((ext_vector_type(2))) float v2f;
typedef __attribute__((ext_vector_type(8))) float v8f;

#define DIM 128
#define KNEG 5
#define WAVES_PER_BLOCK 8
#define BLOCK (WAVES_PER_BLOCK * 32)

// One wave handles 16 batch elements. A = 16 u-rows (16x128 fp32),
// per target t (pos + 5 neg): B = 16 target-rows, 32x V_WMMA_F32_16X16X4_F32,
// keep only the diagonal of the 16x16 fp32 accumulator.
__global__ __launch_bounds__(BLOCK) void sg_wmma_kernel(
    const int* __restrict__ pos_u, const int* __restrict__ pos_v,
    const int* __restrict__ neg_v, const float* __restrict__ uW,
    const float* __restrict__ vW, float* __restrict__ partial, int nGroups)
{
    const int tid    = threadIdx.x;
    const int wave   = tid >> 5;
    const int lane   = tid & 31;
    const int m      = lane & 15;                 // row of A / column of B this lane serves
    const int colOff = (lane < 16) ? 0 : 2;       // K sub-offset per A/B VGPR layout
    const int group  = blockIdx.x * WAVES_PER_BLOCK + wave;

    float posPart = 0.f, negPart = 0.f;

    if (group < nGroups) {                        // wave-uniform branch: EXEC stays all-ones
        const int b = group * 16 + m;             // batch element owned by this lane

        const float* urow = uW + (long long)pos_u[b] * DIM;
        const float* trow[KNEG + 1];
        trow[0] = vW + (long long)pos_v[b] * DIM;
        #pragma unroll
        for (int k = 0; k < KNEG; ++k)
            trow[1 + k] = vW + (long long)neg_v[b * KNEG + k] * DIM;

        __builtin_prefetch(urow, 0, 0);           // global_prefetch_b8
        #pragma unroll
        for (int t = 0; t < KNEG + 1; ++t) __builtin_prefetch(trow[t], 0, 0);

        // A fragments: 32 chunks of K=4; lane = row m, elems (colOff, colOff+1)
        v2f a[DIM / 4];
        #pragma unroll
        for (int c = 0; c < DIM / 4; ++c)
            a[c] = *(const v2f*)(urow + 4 * c + colOff);

        #pragma unroll
        for (int t = 0; t < KNEG + 1; ++t) {
            const float* tr = trow[t];
            v8f acc = {0.f, 0.f, 0.f, 0.f, 0.f, 0.f, 0.f, 0.f};
            #pragma unroll
            for (int c = 0; c < DIM / 4; ++c) {
                v2f bf = *(const v2f*)(tr + 4 * c + colOff);
                acc = __builtin_amdgcn_wmma_f32_16x16x4_f32(
                    false, a[c], false, bf, (short)0, acc, false, false);
            }
            // Diagonal of C: m=0..7 -> lane m, VGPR m ; m=8..15 -> lane m+16, VGPR m-8
            const bool hasDiag = (lane < 8) || (lane >= 24);
            const int  j = lane & 7;
            float dg = acc[0];
            #pragma unroll
            for (int q = 1; q < 8; ++q) dg = (j == q) ? acc[q] : dg;

            if (hasDiag) {
                float s = fminf(fmaxf(dg, -6.f), 6.f);
                if (t == 0) posPart += log1pf(expf(-s));   // -log_sigmoid(s)
                else        negPart += log1pf(expf(s));    // -log_sigmoid(-s)
            }
        }
    }

    // wave32 butterfly reduction
    #pragma unroll
    for (int off = 16; off > 0; off >>= 1) {
        posPart += __shfl_xor(posPart, off, 32);
        negPart += __shfl_xor(negPart, off, 32);
    }

    __shared__ float redP[WAVES_PER_BLOCK], redN[WAVES_PER_BLOCK];
    if (lane == 0) { redP[wave] = posPart; redN[wave] = negPart; }
    __syncthreads();
    if (tid == 0) {
        float p = 0.f, n = 0.f;
        #pragma unroll
        for (int w = 0; w < WAVES_PER_BLOCK; ++w) { p += redP[w]; n += redN[w]; }
        partial[2 * blockIdx.x + 0] = p;
        partial[2 * blockIdx.x + 1] = n;
    }
}

// Deterministic fixed-order final reduction (single block).
__global__ __launch_bounds__(256) void sg_reduce_kernel(
    const float* __restrict__ partial, int n, float* __restrict__ out)
{
    __shared__ float sp[256], sn[256];
    const int tid = threadIdx.x;
    float p = 0.f, q = 0.f;
    for (int i = tid; i < n; i += 256) { p += partial[2 * i]; q += partial[2 * i + 1]; }
    sp[tid] = p; sn[tid] = q;
    __syncthreads();
    for (int s = 128; s > 0; s >>= 1) {
        if (tid < s) { sp[tid] += sp[tid + s]; sn[tid] += sn[tid + s]; }
        __syncthreads();
    }
    if (tid == 0) { out[0] = sp[0]; out[1] = sn[0]; }
}

extern "C" void kernel_launch(void* const* d_in, const int* in_sizes, int n_in,
                              void* d_out, int out_size, void* d_ws, size_t ws_size,
                              hipStream_t stream)
{
    const int*   pos_u = (const int*)  d_in[0];
    const int*   pos_v = (const int*)  d_in[1];
    const int*   neg_v = (const int*)  d_in[2];
    const float* uW    = (const float*)d_in[3];
    const float* vW    = (const float*)d_in[4];
    float* out     = (float*)d_out;
    float* partial = (float*)d_ws;   // 2 floats per block

    const int B       = in_sizes[0];           // 65536 (multiple of 16)
    const int nGroups = B / 16;
    const int nBlocks = (nGroups + WAVES_PER_BLOCK - 1) / WAVES_PER_BLOCK;

    sg_wmma_kernel<<<nBlocks, BLOCK, 0, stream>>>(pos_u, pos_v, neg_v, uW, vW,
                                                  partial, nGroups);
    sg_reduce_kernel<<<1, 256, 0, stream>>>(partial, nBlocks, out);
}